// MLPPolicyMF_10213432230162
// MI455X (gfx1250) — compile-verified
//
#include <hip/hip_runtime.h>

typedef __attribute__((ext_vector_type(2))) float v2f;
typedef __attribute__((ext_vector_type(4))) float v4f;
typedef __attribute__((ext_vector_type(8))) float v8f;

#define K_NN    11
#define NW      8              // waves per workgroup in QEC kernel
#define C_TAB   50000
#define D_DIM   64
#define A_ACT   4
#define B_BATCH 1024
#define CHUNKS  (C_TAB / 16)   // 3125, exact
#define BT      2              // 16-row batch tiles per workgroup (32 obs cols)

// ---------------------------------------------------------------------------
// Kernel 0: precompute ||s||^2 for every table row (A*C rows). One streaming
// pass over the 51.2 MB table; also warms the 192 MB L2 for the main kernel.
// ---------------------------------------------------------------------------
__global__ __launch_bounds__(256) void s2_kernel(
    const float* __restrict__ ts, float* __restrict__ s2, int nrows)
{
  int r = blockIdx.x * 256 + threadIdx.x;
  if (r >= nrows) return;
  const v4f* p = (const v4f*)(ts + (size_t)r * D_DIM);
  float acc = 0.0f;
  #pragma unroll
  for (int i = 0; i < D_DIM / 4; ++i) {
    v4f v = p[i];
    acc += v.x * v.x + v.y * v.y + v.z * v.z + v.w * v.w;
  }
  s2[r] = acc;
}

// ---------------------------------------------------------------------------
// Kernel 1: fused distance-GEMM (fp32 WMMA 16x16x4) + streaming top-11.
// One workgroup = (32 batch rows, 1 action). 8 waves split the 50000
// candidates; per-lane register top-11 lists; LDS merge at the end.
// key = ||s||^2 - 2*dot  (||x||^2 dropped: constant per batch row).
// PRECOMP: read ||s||^2 from workspace (2 b128 loads) instead of computing
// it in the hot loop (saves ~33 VALU + 9 ds_bpermute per chunk).
// ---------------------------------------------------------------------------
template <bool PRECOMP>
__global__ __launch_bounds__(NW * 32) void qec_topk_kernel(
    const float* __restrict__ obs,
    const float* __restrict__ tstates,
    const float* __restrict__ tqvals,
    const float* __restrict__ s2tab,     // [A*C], valid iff PRECOMP
    float* __restrict__ qec_out)
{
  const int lane   = threadIdx.x & 31;
  const int wid    = threadIdx.x >> 5;
  const int lane15 = lane & 15;
  const int hi     = lane >> 4;   // 0 or 1 (lane half)
  const int hi8    = hi << 3;     // C/D row-block offset for this half
  const int koff   = hi << 1;     // A/B K sub-offset (0 or 2)

  const int a  = blockIdx.x & (A_ACT - 1);
  const int b0 = (blockIdx.x >> 2) * (16 * BT);

  const float* __restrict__ ts_a = tstates + (size_t)a * C_TAB * D_DIM;
  const float* __restrict__ tq_a = tqvals  + (size_t)a * C_TAB;
  const float* __restrict__ s2_a = s2tab   + (size_t)a * C_TAB;

  // Preload observation B-fragments for BT batch tiles.
  // Lane l, VGPR v, step s: obs[b0+16t + l%16][4s + 2*(l/16) + v] -> b64/lane.
  v2f bf[BT][16];
  #pragma unroll
  for (int t = 0; t < BT; ++t) {
    const float* orow = obs + (size_t)(b0 + 16 * t + lane15) * D_DIM + koff;
    #pragma unroll
    for (int s = 0; s < 16; ++s)
      bf[t][s] = *(const v2f*)(orow + 4 * s);
  }

  // Per-lane running top-K per batch tile (sorted ascending), in registers.
  float lk[BT][K_NN], lq[BT][K_NN], kmax[BT];
  #pragma unroll
  for (int t = 0; t < BT; ++t) {
    #pragma unroll
    for (int i = 0; i < K_NN; ++i) { lk[t][i] = 3.0e38f; lq[t][i] = 0.0f; }
    kmax[t] = 3.0e38f;
  }

  for (int j = wid; j < CHUNKS; j += NW) {
    const int cbase = j << 4;

    // A-fragments: 16 candidate rows x K=64 fp32 (lane halves hold K%4 pairs).
    const float* arow = ts_a + (size_t)(cbase + lane15) * D_DIM + koff;
    v2f af[16];
    #pragma unroll
    for (int s = 0; s < 16; ++s)
      af[s] = *(const v2f*)(arow + 4 * s);

    // Cover latency of this wave's next chunk (NW*16 rows ahead).
    if (j + NW < CHUNKS)
      __builtin_prefetch(arow + (size_t)NW * 16 * D_DIM, 0, 1);

    float s2row[8];
    if (PRECOMP) {
      v4f sa = *(const v4f*)(s2_a + cbase + hi8);
      v4f sb = *(const v4f*)(s2_a + cbase + hi8 + 4);
      s2row[0] = sa.x; s2row[1] = sa.y; s2row[2] = sa.z; s2row[3] = sa.w;
      s2row[4] = sb.x; s2row[5] = sb.y; s2row[6] = sb.z; s2row[7] = sb.w;
    }

    v8f acc[BT];
    #pragma unroll
    for (int t = 0; t < BT; ++t) acc[t] = (v8f){};

    float s2p = 0.0f;
    #pragma unroll
    for (int s = 0; s < 16; ++s) {
      #pragma unroll
      for (int t = 0; t < BT; ++t)
        acc[t] = __builtin_amdgcn_wmma_f32_16x16x4_f32(
            false, af[s], false, bf[t][s], (short)0, acc[t], false, false);
      if (!PRECOMP)
        s2p += af[s].x * af[s].x + af[s].y * af[s].y;   // partial ||s||^2
    }
    if (!PRECOMP) {
      // Lanes l and l+16 hold complementary halves of candidate row l%16.
      float s2full = s2p + __shfl_xor(s2p, 16, 32);
      #pragma unroll
      for (int r = 0; r < 8; ++r)
        s2row[r] = __shfl(s2full, hi8 + r, 32);
    }

    #pragma unroll
    for (int r = 0; r < 8; ++r) {
      // C/D layout: lane l, VGPR r -> candidate M = r + 8*(l/16), col N = l%16
      #pragma unroll
      for (int t = 0; t < BT; ++t) {
        float key = s2row[r] - 2.0f * acc[t][r];
        if (key < kmax[t]) {                            // execz-skipped when cold
          float cq = tq_a[cbase + hi8 + r];
          float ck = key;
          #pragma unroll
          for (int i = 0; i < K_NN; ++i) {              // branch-free bubble insert
            bool  sw = ck < lk[t][i];
            float tk = lk[t][i], tq = lq[t][i];
            lk[t][i] = sw ? ck : tk;
            lq[t][i] = sw ? cq : tq;
            ck       = sw ? tk : ck;
            cq       = sw ? tq : cq;
          }
          kmax[t] = lk[t][K_NN - 1];
        }
      }
    }
  }

  // ------------------- cross-wave / cross-half merge via LDS ----------------
  __shared__ float sk[NW * BT * 32 * K_NN];
  __shared__ float sq[NW * BT * 32 * K_NN];
  #pragma unroll
  for (int t = 0; t < BT; ++t) {
    int base = ((wid * BT + t) * 32 + lane) * K_NN;
    #pragma unroll
    for (int i = 0; i < K_NN; ++i) { sk[base + i] = lk[t][i]; sq[base + i] = lq[t][i]; }
  }
  __syncthreads();

  if (threadIdx.x < 16 * BT) {
    const int n    = threadIdx.x;                       // output col b0+n
    const int tile = n >> 4;
    const int col  = n & 15;
    float mk[K_NN], mq[K_NN];
    #pragma unroll
    for (int i = 0; i < K_NN; ++i) { mk[i] = 3.0e38f; mq[i] = 0.0f; }
    float mmax = 3.0e38f;

    for (int w = 0; w < NW; ++w) {                      // 8 waves x 2 lane-halves
      for (int h = 0; h < 2; ++h) {
        int lbase = ((w * BT + tile) * 32 + h * 16 + col) * K_NN;
        for (int i = 0; i < K_NN; ++i) {
          float ck = sk[lbase + i];
          if (ck >= mmax) break;                        // source lists are sorted
          float cq = sq[lbase + i];
          #pragma unroll
          for (int u = 0; u < K_NN; ++u) {
            bool  sw = ck < mk[u];
            float tk = mk[u], tq = mq[u];
            mk[u] = sw ? ck : tk;
            mq[u] = sw ? cq : tq;
            ck    = sw ? tk : ck;
            cq    = sw ? tq : cq;
          }
          mmax = mk[K_NN - 1];
        }
      }
    }
    float qsum = 0.0f;
    #pragma unroll
    for (int i = 0; i < K_NN; ++i) qsum += mq[i];
    qec_out[(size_t)(b0 + n) * A_ACT + a] = qsum * (1.0f / K_NN);
  }
}

// ---------------------------------------------------------------------------
// Kernel 2: tiny MLP (64->64->64->4) + 0.5*(qec+qnet) + argmax.
// One thread per batch row; activations in padded LDS rows (no bank
// conflicts); weight reads are wave-uniform -> scalar loads.
// ---------------------------------------------------------------------------
#define MLP_T 64
__global__ __launch_bounds__(MLP_T) void mlp_combine_kernel(
    const float* __restrict__ obs,
    const float* __restrict__ qec,
    const float* __restrict__ W1, const float* __restrict__ b1,
    const float* __restrict__ W2, const float* __restrict__ b2,
    const float* __restrict__ W3, const float* __restrict__ b3,
    float* __restrict__ out)
{
  __shared__ float bufA[MLP_T][D_DIM + 1];
  __shared__ float bufB[MLP_T][D_DIM + 1];
  const int t = threadIdx.x;
  const int b = blockIdx.x * MLP_T + t;

  const float* orow = obs + (size_t)b * D_DIM;
  for (int d = 0; d < D_DIM; ++d) bufA[t][d] = orow[d];

  for (int jj = 0; jj < D_DIM; ++jj) {
    float acc = b1[jj];
    for (int d = 0; d < D_DIM; ++d) acc = fmaf(bufA[t][d], W1[d * D_DIM + jj], acc);
    bufB[t][jj] = fmaxf(acc, 0.0f);
  }
  for (int jj = 0; jj < D_DIM; ++jj) {
    float acc = b2[jj];
    for (int d = 0; d < D_DIM; ++d) acc = fmaf(bufB[t][d], W2[d * D_DIM + jj], acc);
    bufA[t][jj] = fmaxf(acc, 0.0f);
  }

  float q[A_ACT];
  #pragma unroll
  for (int jj = 0; jj < A_ACT; ++jj) {
    float acc = b3[jj];
    for (int d = 0; d < D_DIM; ++d) acc = fmaf(bufA[t][d], W3[d * A_ACT + jj], acc);
    q[jj] = 0.5f * (acc + qec[(size_t)b * A_ACT + jj]);
  }

  int amax = 0; float best = q[0];
  #pragma unroll
  for (int jj = 1; jj < A_ACT; ++jj)
    if (q[jj] > best) { best = q[jj]; amax = jj; }     // first-max tie-break

  out[b] = (float)amax;                                // action, then q (B x A)
  float* oq = out + B_BATCH;
  #pragma unroll
  for (int jj = 0; jj < A_ACT; ++jj) oq[(size_t)b * A_ACT + jj] = q[jj];
}

// ---------------------------------------------------------------------------
extern "C" void kernel_launch(void* const* d_in, const int* in_sizes, int n_in,
                              void* d_out, int out_size, void* d_ws, size_t ws_size,
                              hipStream_t stream) {
  const float* obs = (const float*)d_in[0];
  const float* ts  = (const float*)d_in[1];
  const float* tq  = (const float*)d_in[2];
  const float* W1  = (const float*)d_in[3];
  const float* b1  = (const float*)d_in[4];
  const float* W2  = (const float*)d_in[5];
  const float* b2  = (const float*)d_in[6];
  const float* W3  = (const float*)d_in[7];
  const float* b3  = (const float*)d_in[8];

  float* out    = (float*)d_out;
  float* qec_ws = (float*)d_ws;                        // B_BATCH * A_ACT floats
  float* s2_ws  = qec_ws + (size_t)B_BATCH * A_ACT;    // A_ACT * C_TAB floats

  const size_t need = ((size_t)B_BATCH * A_ACT + (size_t)A_ACT * C_TAB) * sizeof(float);
  const dim3 g1((B_BATCH / (16 * BT)) * A_ACT);        // 128 workgroups

  if (ws_size >= need) {
    const int nrows = A_ACT * C_TAB;
    s2_kernel<<<(nrows + 255) / 256, 256, 0, stream>>>(ts, s2_ws, nrows);
    qec_topk_kernel<true><<<g1, NW * 32, 0, stream>>>(obs, ts, tq, s2_ws, qec_ws);
  } else {
    qec_topk_kernel<false><<<g1, NW * 32, 0, stream>>>(obs, ts, tq, qec_ws, qec_ws);
  }

  mlp_combine_kernel<<<B_BATCH / MLP_T, MLP_T, 0, stream>>>(obs, qec_ws,
                                                            W1, b1, W2, b2, W3, b3, out);
}